// LSTM_Model_36017595744657
// MI455X (gfx1250) — compile-verified
//
#include <hip/hip_runtime.h>
#include <math.h>

typedef __attribute__((ext_vector_type(2))) float v2f;
typedef __attribute__((ext_vector_type(8))) float v8f;

#define T_STEPS 160
#define BATCH   640
#define FDIM    40
#define HDIM    768
#define PJDIM   256
#define FOURH   3072

// ---------------------------------------------------------------------------
// Gate GEMM + cell update.
//   z = [inp, h_prev] @ W + b   (gate order i, j, f, o along 4H)
//   c_new = sig(f+1)*c + sig(i)*tanh(j);  m = sig(o)*tanh(c_new)
// grid = (BATCH/32, HDIM/64), block = 256 (8 waves of 32).
// Wave w: rows 16*(w&1), gate g = w>>2? no: g = w>>1, 4 col-subtiles of 16.
// ---------------------------------------------------------------------------
__global__ __launch_bounds__(256)
void lstm_gate_cell(const float* __restrict__ inp, int Kin,
                    const float* __restrict__ hprev,
                    const float* __restrict__ W,     // [(Kin+PJ) x 4H] row-major
                    const float* __restrict__ bias,  // [4H]
                    float* __restrict__ c,           // [B x H] in/out
                    float* __restrict__ m)           // [B x H] out
{
    __shared__ float smem[32 * 513];   // A-stage; reused as z-buffer afterwards

    const int tid  = threadIdx.x;
    const int wave = tid >> 5;
    const int lane = tid & 31;
    const int lo   = lane & 15;
    const int hi   = lane >> 4;
    const int r_base = blockIdx.x * 32;
    const int h_base = blockIdx.y * 64;
    const int K = Kin + PJDIM;

    // Stage A = concat(inp, h_prev) for 32 rows into LDS (stride 513: conflict-free)
    for (int idx = tid; idx < 32 * K; idx += 256) {
        int r = idx / K;
        int k = idx - r * K;
        float v = (k < Kin) ? inp[(size_t)(r_base + r) * Kin + k]
                            : hprev[(size_t)(r_base + r) * PJDIM + (k - Kin)];
        smem[r * 513 + k] = v;
    }
    __syncthreads();

    const int r0 = (wave & 1) * 16;     // 16-row subtile
    const int g  = wave >> 1;           // gate 0..3 (i,j,f,o)
    const int ncolbase = g * HDIM + h_base;

    v8f zero = {0.f,0.f,0.f,0.f,0.f,0.f,0.f,0.f};
    v8f acc[4] = {zero, zero, zero, zero};

    for (int k0 = 0; k0 < K; k0 += 4) {
        const int ka = k0 + 2 * hi;     // lanes 16-31 hold K+2 / K+3
        v2f a;
        a.x = smem[(r0 + lo) * 513 + ka];
        a.y = smem[(r0 + lo) * 513 + ka + 1];
#pragma unroll
        for (int j = 0; j < 4; ++j) {
            const int ncol = ncolbase + j * 16 + lo;
            v2f b;
            b.x = W[(size_t)ka * FOURH + ncol];
            b.y = W[(size_t)(ka + 1) * FOURH + ncol];
            acc[j] = __builtin_amdgcn_wmma_f32_16x16x4_f32(
                false, a, false, b, (short)0, acc[j], false, false);
        }
    }
    __syncthreads();   // all waves done reading staged A

    // Scatter z tiles into LDS as z[row(32)][col(256)], stride 260.
    // col = g*64 + u maps to global gate column g*768 + h_base + u.
#pragma unroll
    for (int j = 0; j < 4; ++j) {
        const int cb = g * 64 + j * 16 + lo;
#pragma unroll
        for (int v = 0; v < 8; ++v) {
            const int r = r0 + v + 8 * hi;   // C/D layout: VGPR v -> M=v / v+8
            smem[r * 260 + cb] = acc[j][v];
        }
    }
    __syncthreads();

    // Fused elementwise cell update (each thread: 8 (row,u) pairs)
    for (int e = tid; e < 32 * 64; e += 256) {
        const int r = e >> 6;
        const int u = e & 63;
        const int hcol = h_base + u;
        float zi = smem[r * 260 +   0 + u] + bias[0 * HDIM + hcol];
        float zj = smem[r * 260 +  64 + u] + bias[1 * HDIM + hcol];
        float zf = smem[r * 260 + 128 + u] + bias[2 * HDIM + hcol];
        float zo = smem[r * 260 + 192 + u] + bias[3 * HDIM + hcol];
        const size_t ci = (size_t)(r_base + r) * HDIM + hcol;
        float cold = c[ci];
        float si = 1.f / (1.f + expf(-zi));
        float sf = 1.f / (1.f + expf(-(zf + 1.0f)));   // FORGET_BIAS = 1
        float so = 1.f / (1.f + expf(-zo));
        float cn = sf * cold + si * tanhf(zj);
        c[ci] = cn;
        m[ci] = so * tanhf(cn);
    }
}

// ---------------------------------------------------------------------------
// Projection: h = m @ P, [640,768] @ [768,256].
// grid = (BATCH/32, PJ/64), block = 256; one 16x16 WMMA tile per wave.
// ---------------------------------------------------------------------------
__global__ __launch_bounds__(256)
void lstm_proj(const float* __restrict__ m,
               const float* __restrict__ P,   // [H x PJ] row-major
               float* __restrict__ h)         // [B x PJ]
{
    const int tid  = threadIdx.x;
    const int wave = tid >> 5;
    const int lane = tid & 31;
    const int lo   = lane & 15;
    const int hi   = lane >> 4;
    const int r_base = blockIdx.x * 32 + (wave & 1) * 16;
    const int c_base = blockIdx.y * 64 + (wave >> 1) * 16;

    v8f acc = {0.f,0.f,0.f,0.f,0.f,0.f,0.f,0.f};
    const float* mrow = m + (size_t)(r_base + lo) * HDIM;

    for (int k0 = 0; k0 < HDIM; k0 += 4) {
        const int ka = k0 + 2 * hi;
        v2f a;
        a.x = mrow[ka];
        a.y = mrow[ka + 1];
        v2f b;
        b.x = P[(size_t)ka * PJDIM + c_base + lo];
        b.y = P[(size_t)(ka + 1) * PJDIM + c_base + lo];
        acc = __builtin_amdgcn_wmma_f32_16x16x4_f32(
            false, a, false, b, (short)0, acc, false, false);
    }
#pragma unroll
    for (int v = 0; v < 8; ++v)
        h[(size_t)(r_base + v + 8 * hi) * PJDIM + c_base + lo] = acc[v];
}

// ---------------------------------------------------------------------------
// out[b,:] = h2[b,:] / sqrt(max(sum(h2[b,:]^2), 1e-12))
// ---------------------------------------------------------------------------
__global__ __launch_bounds__(256)
void l2norm_kernel(const float* __restrict__ hfin, float* __restrict__ out)
{
    __shared__ float red[256];
    const int b = blockIdx.x, tid = threadIdx.x;
    float v = hfin[(size_t)b * PJDIM + tid];
    red[tid] = v * v;
    __syncthreads();
    for (int s = 128; s > 0; s >>= 1) {
        if (tid < s) red[tid] += red[tid + s];
        __syncthreads();
    }
    const float inv = rsqrtf(fmaxf(red[0], 1e-12f));
    out[(size_t)b * PJDIM + tid] = v * inv;
}

// ---------------------------------------------------------------------------
extern "C" void kernel_launch(void* const* d_in, const int* in_sizes, int n_in,
                              void* d_out, int out_size, void* d_ws, size_t ws_size,
                              hipStream_t stream)
{
    const float* x  = (const float*)d_in[0];
    const float* Ws[3] = { (const float*)d_in[1], (const float*)d_in[4], (const float*)d_in[7] };
    const float* bs[3] = { (const float*)d_in[2], (const float*)d_in[5], (const float*)d_in[8] };
    const float* Ps[3] = { (const float*)d_in[3], (const float*)d_in[6], (const float*)d_in[9] };
    float* out = (float*)d_out;

    float* ws = (float*)d_ws;
    const size_t cN = (size_t)BATCH * HDIM;    // 491520
    const size_t hN = (size_t)BATCH * PJDIM;   // 163840
    float* cs[3] = { ws, ws + cN, ws + 2 * cN };
    float* hs[3] = { ws + 3 * cN, ws + 3 * cN + hN, ws + 3 * cN + 2 * hN };
    float* mbuf  = ws + 3 * cN + 3 * hN;

    // zero initial c and h states every launch (deterministic; capture-safe)
    hipMemsetAsync(ws, 0, (3 * cN + 3 * hN) * sizeof(float), stream);

    const dim3 gGate(BATCH / 32, HDIM / 64);
    const dim3 gProj(BATCH / 32, PJDIM / 64);

    for (int t = 0; t < T_STEPS; ++t) {
        for (int l = 0; l < 3; ++l) {
            const float* inp = (l == 0) ? (x + (size_t)t * BATCH * FDIM) : hs[l - 1];
            const int Kin = (l == 0) ? FDIM : PJDIM;
            lstm_gate_cell<<<gGate, 256, 0, stream>>>(inp, Kin, hs[l], Ws[l], bs[l], cs[l], mbuf);
            lstm_proj<<<gProj, 256, 0, stream>>>(mbuf, Ps[l], hs[l]);
        }
    }
    l2norm_kernel<<<BATCH, 256, 0, stream>>>(hs[2], out);
}